// NMRGNN_80951543595698
// MI455X (gfx1250) — compile-verified
//
#include <hip/hip_runtime.h>
#include <hip/hip_bf16.h>
#include <cstddef>
#include <cstdint>

#define N_NODES 100000
#define N_EDGES 3200000
#define BN_EPS 1e-5f

typedef __attribute__((ext_vector_type(2))) float v2f;
typedef __attribute__((ext_vector_type(8))) float v8f;

// D = A(16x4 f32) * B(4x16 f32) + C(16x16 f32), wave32 WMMA
__device__ __forceinline__ v8f wmma_f32(v2f a, v2f b, v8f c) {
  return __builtin_amdgcn_wmma_f32_16x16x4_f32(false, a, false, b, (short)0, c,
                                               false, false);
}

// ---------------------------------------------------------------- zero (f4)
__global__ void zero_f4_kernel(float4* __restrict__ p, long long n4) {
  long long i = (long long)blockIdx.x * blockDim.x + threadIdx.x;
  if (i < n4) p[i] = make_float4(0.f, 0.f, 0.f, 0.f);
}

// ------------------------------------------------- h = [x | emb[labels] | 0]
__global__ void build_input_kernel(const float* __restrict__ x,
                                   const int* __restrict__ labels,
                                   const float* __restrict__ emb,
                                   float* __restrict__ h) {
  long long idx = (long long)blockIdx.x * blockDim.x + threadIdx.x;
  if (idx >= (long long)N_NODES * 16) return;
  const int i = (int)(idx >> 4);
  const int c = (int)(idx & 15);
  float v = 0.0f;
  if (c < 2)       v = x[(size_t)i * 2 + c];
  else if (c < 10) v = emb[(size_t)labels[i] * 8 + (c - 2)];
  h[(size_t)i * 64 + c] = v;
}

// ------------------------------------------------------ pad W1_0 (10x64->16x64)
__global__ void pad_w1_kernel(const float* __restrict__ W1,
                              float* __restrict__ Wpad) {
  const int idx = blockIdx.x * blockDim.x + threadIdx.x;
  if (idx >= 16 * 64) return;
  const int r = idx >> 6;
  Wpad[idx] = (r < 10) ? W1[idx] : 0.0f;
}

// ----------------------------------------------- agg[dst] += h[src] (scatter)
__global__ void scatter_kernel(const float* __restrict__ h,
                               const int* __restrict__ edge_index,
                               float* __restrict__ agg, int log2w) {
  const long long idx = (long long)blockIdx.x * blockDim.x + threadIdx.x;
  const long long total = (long long)N_EDGES << log2w;
  if (idx >= total) return;
  const int e = (int)(idx >> log2w);
  const int c = (int)(idx & ((1 << log2w) - 1));
  const int s = edge_index[e];
  const int d = edge_index[N_EDGES + e];
  const float v = h[(size_t)s * 64 + c];
  unsafeAtomicAdd(&agg[(size_t)d * 64 + c], v);
}

// ---- fused GIN combine + 2-layer MLP (WMMA f32) + BN partial sums ----------
// one wave per 16-row tile; 4 waves / block
__global__ void __launch_bounds__(128)
gin_mlp_kernel(float* __restrict__ h, const float* __restrict__ agg,
               const float* __restrict__ W1, const float* __restrict__ b1,
               const float* __restrict__ W2, const float* __restrict__ b2,
               const float* __restrict__ eps_ptr, int eps_idx, int K,
               float* __restrict__ stats) {
  __shared__ float ytile[4 * 16 * 68];  // stride 68 -> conflict-free transpose
  __shared__ float sstat[128];
  const int tid   = threadIdx.x;
  const int wave  = tid >> 5;
  const int lane  = tid & 31;
  const int lrow  = lane & 15;
  const int lhalf = lane >> 4;
  const int tile  = blockIdx.x * 4 + wave;

  if (tid < 128) sstat[tid] = 0.0f;
  __syncthreads();

  if (tile < (N_NODES / 16)) {
    const int base = tile * 16;
    const float onepe = 1.0f + eps_ptr[eps_idx];
    const int koff = lhalf * 2;

    // GEMM1: y1 = ((1+eps)*h + agg) @ W1   (16xK * Kx64)
    v8f acc[4] = {};
    for (int kk = 0; kk < K; kk += 4) {
      const int k0 = kk + koff;
      const float* hp = h   + (size_t)(base + lrow) * 64 + k0;
      const float* ap = agg + (size_t)(base + lrow) * 64 + k0;
      v2f afrag;
      afrag.x = onepe * hp[0] + ap[0];
      afrag.y = onepe * hp[1] + ap[1];
      const float* wp = W1 + (size_t)k0 * 64 + lrow;
#pragma unroll
      for (int nt = 0; nt < 4; ++nt) {
        v2f bfrag;
        bfrag.x = wp[nt * 16];
        bfrag.y = wp[nt * 16 + 64];
        acc[nt] = wmma_f32(afrag, bfrag, acc[nt]);
      }
    }

    // bias + ReLU, stage to LDS for the cross-lane transpose (C-layout -> A-layout)
    float* my = ytile + wave * (16 * 68);
#pragma unroll
    for (int nt = 0; nt < 4; ++nt) {
      const float bv = b1[nt * 16 + lrow];
#pragma unroll
      for (int r = 0; r < 8; ++r)
        my[(r + lhalf * 8) * 68 + nt * 16 + lrow] = fmaxf(acc[nt][r] + bv, 0.f);
    }

    // GEMM2: y2 = y1 @ W2   (16x64 * 64x64)
    v8f acc2[4] = {};
    for (int kk = 0; kk < 64; kk += 4) {
      const int k0 = kk + koff;
      v2f afrag;
      afrag.x = my[lrow * 68 + k0];
      afrag.y = my[lrow * 68 + k0 + 1];
      const float* wp = W2 + (size_t)k0 * 64 + lrow;
#pragma unroll
      for (int nt = 0; nt < 4; ++nt) {
        v2f bfrag;
        bfrag.x = wp[nt * 16];
        bfrag.y = wp[nt * 16 + 64];
        acc2[nt] = wmma_f32(afrag, bfrag, acc2[nt]);
      }
    }

    // bias + ReLU, in-place store, BN partial sums
#pragma unroll
    for (int nt = 0; nt < 4; ++nt) {
      const float bv = b2[nt * 16 + lrow];
      float s = 0.0f, s2 = 0.0f;
#pragma unroll
      for (int r = 0; r < 8; ++r) {
        const float v = fmaxf(acc2[nt][r] + bv, 0.f);
        h[(size_t)(base + r + lhalf * 8) * 64 + nt * 16 + lrow] = v;
        s += v;
        s2 += v * v;
      }
      unsafeAtomicAdd(&sstat[nt * 16 + lrow], s);
      unsafeAtomicAdd(&sstat[64 + nt * 16 + lrow], s2);
    }
  }
  __syncthreads();
  if (tid < 128) unsafeAtomicAdd(&stats[tid], sstat[tid]);
}

// ------------------------------------------ fold BN into scale/shift per col
__global__ void bn_finalize_kernel(const float* __restrict__ stats,
                                   const float* __restrict__ gamma,
                                   const float* __restrict__ beta,
                                   float* __restrict__ ss) {
  const int c = threadIdx.x;
  if (c >= 64) return;
  const float inv_n = 1.0f / (float)N_NODES;
  const float mu  = stats[c] * inv_n;
  const float var = stats[64 + c] * inv_n - mu * mu;
  const float rs  = rsqrtf(var + BN_EPS);
  const float sc  = gamma[c] * rs;
  ss[c]      = sc;
  ss[64 + c] = beta[c] - mu * sc;
}

// ---------------------------------------------- h = relu(h*scale + shift), f4
__global__ void bn_relu_kernel(float* __restrict__ h,
                               const float* __restrict__ ss) {
  const long long i = (long long)blockIdx.x * blockDim.x + threadIdx.x;
  if (i >= (long long)N_NODES * 16) return;
  const int c4 = (int)(i & 15) * 4;
  float4 v = ((const float4*)h)[i];
  v.x = fmaxf(v.x * ss[c4 + 0] + ss[64 + c4 + 0], 0.f);
  v.y = fmaxf(v.y * ss[c4 + 1] + ss[64 + c4 + 1], 0.f);
  v.z = fmaxf(v.z * ss[c4 + 2] + ss[64 + c4 + 2], 0.f);
  v.w = fmaxf(v.w * ss[c4 + 3] + ss[64 + c4 + 3], 0.f);
  ((float4*)h)[i] = v;
}

// ---------------------------------------------------- out = h @ Wf + bf
__global__ void final_kernel(const float* __restrict__ h,
                             const float* __restrict__ Wf,
                             const float* __restrict__ bf,
                             float* __restrict__ out) {
  const int i = blockIdx.x * blockDim.x + threadIdx.x;
  if (i >= N_NODES) return;
  const float4* hp = (const float4*)(h + (size_t)i * 64);
  const float4* wp = (const float4*)Wf;
  float s = bf[0];
#pragma unroll
  for (int j = 0; j < 16; ++j) {
    float4 a = hp[j], b = wp[j];
    s += a.x * b.x + a.y * b.y + a.z * b.z + a.w * b.w;
  }
  out[i] = s;
}

extern "C" void kernel_launch(void* const* d_in, const int* in_sizes, int n_in,
                              void* d_out, int out_size, void* d_ws,
                              size_t ws_size, hipStream_t stream) {
  const float* x      = (const float*)d_in[0];
  const int*   labels = (const int*)d_in[1];
  const int*   ei     = (const int*)d_in[2];
  const float* emb    = (const float*)d_in[3];
  const float* W1_0   = (const float*)d_in[4];
  const float* b1_0   = (const float*)d_in[5];
  const float* W2_0   = (const float*)d_in[6];
  const float* b2_0   = (const float*)d_in[7];
  const float* eps_0  = (const float*)d_in[8];
  const float* g_0    = (const float*)d_in[9];
  const float* be_0   = (const float*)d_in[10];
  const float* W1_s   = (const float*)d_in[11];
  const float* b1_s   = (const float*)d_in[12];
  const float* W2_s   = (const float*)d_in[13];
  const float* b2_s   = (const float*)d_in[14];
  const float* eps_s  = (const float*)d_in[15];
  const float* g_s    = (const float*)d_in[16];
  const float* be_s   = (const float*)d_in[17];
  const float* Wf     = (const float*)d_in[18];
  const float* bf     = (const float*)d_in[19];
  float* out = (float*)d_out;

  // workspace layout: agg[N*64] | stats[128] | ss[128] | Wpad[1024] | h[N*64]
  float* agg   = (float*)d_ws;
  float* stats = agg + (size_t)N_NODES * 64;
  float* ss    = stats + 128;
  float* Wpad  = ss + 128;
  float* h     = Wpad + 1024;

  const int ntiles     = N_NODES / 16;        // 6250 (exact)
  const int mlp_blocks = (ntiles + 3) / 4;    // 4 tiles (waves) per block

  build_input_kernel<<<(N_NODES * 16 + 255) / 256, 256, 0, stream>>>(x, labels,
                                                                     emb, h);
  pad_w1_kernel<<<4, 256, 0, stream>>>(W1_0, Wpad);

  // zero agg + stats (+ss, harmless) in one pass
  const long long zero4 = ((long long)N_NODES * 64 + 256) / 4;
  const int zblocks = (int)((zero4 + 255) / 256);

  for (int layer = 0; layer < 3; ++layer) {
    zero_f4_kernel<<<zblocks, 256, 0, stream>>>((float4*)agg, zero4);

    const int log2w = (layer == 0) ? 4 : 6;  // layer0 features live in cols 0..15
    const long long tot = (long long)N_EDGES << log2w;
    scatter_kernel<<<(int)((tot + 255) / 256), 256, 0, stream>>>(h, ei, agg,
                                                                 log2w);

    const float* W1 = (layer == 0) ? Wpad : (W1_s + (size_t)(layer - 1) * 4096);
    const float* b1 = (layer == 0) ? b1_0 : (b1_s + (size_t)(layer - 1) * 64);
    const float* W2 = (layer == 0) ? W2_0 : (W2_s + (size_t)(layer - 1) * 4096);
    const float* b2 = (layer == 0) ? b2_0 : (b2_s + (size_t)(layer - 1) * 64);
    const float* ep = (layer == 0) ? eps_0 : eps_s;
    const int    ei_idx = (layer == 0) ? 0 : (layer - 1);
    const float* gm = (layer == 0) ? g_0 : (g_s + (size_t)(layer - 1) * 64);
    const float* bt = (layer == 0) ? be_0 : (be_s + (size_t)(layer - 1) * 64);
    const int    K  = (layer == 0) ? 16 : 64;

    gin_mlp_kernel<<<mlp_blocks, 128, 0, stream>>>(h, agg, W1, b1, W2, b2, ep,
                                                   ei_idx, K, stats);
    bn_finalize_kernel<<<1, 64, 0, stream>>>(stats, gm, bt, ss);
    bn_relu_kernel<<<(N_NODES * 16 + 255) / 256, 256, 0, stream>>>(h, ss);
  }

  final_kernel<<<(N_NODES + 255) / 256, 256, 0, stream>>>(h, Wf, bf, out);
}